// AdjacencyConv_6227702579797
// MI455X (gfx1250) — compile-verified
//
#include <hip/hip_runtime.h>

typedef float v2f __attribute__((ext_vector_type(2)));
typedef float v4f __attribute__((ext_vector_type(4)));
typedef float v8f __attribute__((ext_vector_type(8)));

#define D 128
#define BN_EPS 1e-5f

// ---------------------------------------------------------------------------
// init: agg = (1+eps)*x ; zero the BN stats accumulators (needed every call)
// ---------------------------------------------------------------------------
__global__ void init_kernel(const float* __restrict__ x,
                            const float* __restrict__ eps,
                            float* __restrict__ agg,
                            float* __restrict__ stats, int n) {
    int i = blockIdx.x * blockDim.x + threadIdx.x;
    float s = 1.0f + eps[0];
    if (i < n) agg[i] = s * x[i];
    if (i < 1024) stats[i] = 0.0f;
}

// ---------------------------------------------------------------------------
// Generic fused GEMM:  out[M,128] = pre(A[M,128]) @ W'[128,128] + bias
//   W row-major with row stride ldw, column offset woff (B[k][n] = W[n*ldw+woff+k])
//   PRE   (compile-time): per-column affine+ReLU on A:  relu(A*s_vec + c_vec)
//   STATS (compile-time): per-column sum / sum-of-squares accumulation (BN)
// One wave -> 5 M-subtiles (16 rows) x one 16-col tile; block = 8 waves
// covering all 128 cols.  V_WMMA_F32_16X16X4_F32 (native fp32 WMMA).
// EXEC stays all-1s: M is a multiple of 80 at every call site.
// Fragment layouts (ISA 7.12.2):
//   A 16x4 : lanes 0-15 hold K={k,k+1} of row M=lane; lanes 16-31 K={k+2,k+3}
//   B 4x16 : VGPR pair holds B[k..k+1][col] (lo lanes), B[k+2..k+3][col] (hi)
//   C/D    : VGPR r -> row r (lo lanes) / row r+8 (hi lanes), col = lane&15
// ---------------------------------------------------------------------------
template <bool PRE, bool STATS>
__global__ __launch_bounds__(256) void gemm_kernel(
    const float* __restrict__ A, const float* __restrict__ W,
    int ldw, int woff,
    const float* __restrict__ bias,
    const float* __restrict__ s_vec, const float* __restrict__ c_vec,
    float* __restrict__ out,
    float* __restrict__ colsum, float* __restrict__ colsumsq) {

    const int lane = threadIdx.x & 31;
    const int wave = threadIdx.x >> 5;
    const int lo   = lane & 15;
    const int hi   = lane >> 4;          // which K-pair / which row-half
    const int col  = wave * 16 + lo;     // output column owned by this lane
    const long rbase = (long)blockIdx.x * 80;

    v8f acc[5];
#pragma unroll
    for (int t = 0; t < 5; ++t)
#pragma unroll
        for (int r = 0; r < 8; ++r) acc[t][r] = 0.0f;

    // per-lane base pointers (K advances via immediate offsets)
    const float* Wp = W + woff + (long)col * ldw + 2 * hi;
    const float* Ap = A + (rbase + lo) * D + 2 * hi;

#pragma unroll 4
    for (int k0 = 0; k0 < D; k0 += 4) {
        // B fragment: lanes hold B[kk][col], B[kk+1][col]
        v2f b = *(const v2f*)(Wp + k0);
        v2f sv, cv;
        if (PRE) {
            sv = *(const v2f*)(s_vec + k0 + 2 * hi);
            cv = *(const v2f*)(c_vec + k0 + 2 * hi);
        }
#pragma unroll
        for (int t = 0; t < 5; ++t) {
            v2f a = *(const v2f*)(Ap + (long)t * 16 * D + k0);
            if (PRE) {  // fused BN affine + ReLU on the A operand
                a.x = fmaxf(0.0f, a.x * sv.x + cv.x);
                a.y = fmaxf(0.0f, a.y * sv.y + cv.y);
            }
            acc[t] = __builtin_amdgcn_wmma_f32_16x16x4_f32(
                false, a, false, b, (short)0, acc[t], false, false);
        }
    }

    const float bv = bias ? bias[col] : 0.0f;
    float bsum = 0.0f, bsq = 0.0f;
#pragma unroll
    for (int t = 0; t < 5; ++t) {
        const long rowb = rbase + t * 16 + 8 * hi;  // C/D layout: hi lanes = rows+8
#pragma unroll
        for (int r = 0; r < 8; ++r) {
            float v = acc[t][r] + bv;
            out[(rowb + r) * D + col] = v;
            if (STATS) { bsum += v; bsq += v * v; }
        }
    }
    if (STATS) {
        unsafeAtomicAdd(&colsum[col], bsum);
        unsafeAtomicAdd(&colsumsq[col], bsq);
    }
}

// ---------------------------------------------------------------------------
// Edge phase: msg = relu(xW[src] + eW[bridge]) scatter-added into agg[dst].
// One 32-lane wave per edge, 4 floats per lane; everything L2-resident.
// ---------------------------------------------------------------------------
__global__ __launch_bounds__(256) void edge_kernel(
    const float* __restrict__ xW, const float* __restrict__ eW,
    const int* __restrict__ src, const int* __restrict__ dst,
    const int* __restrict__ bridge, float* __restrict__ agg, int E) {
    long tid = (long)blockIdx.x * blockDim.x + threadIdx.x;
    int e = (int)(tid >> 5);
    int lane = (int)(tid & 31);
    if (e >= E) return;
    int s = src[e], d = dst[e], b = bridge[e];
    v4f xv = *(const v4f*)(xW + (long)s * D + lane * 4);
    v4f ev = *(const v4f*)(eW + (long)b * D + lane * 4);
    float* ap = agg + (long)d * D + lane * 4;
#pragma unroll
    for (int i = 0; i < 4; ++i) {
        float m = fmaxf(0.0f, xv[i] + ev[i]);
        unsafeAtomicAdd(ap + i, m);  // global_atomic_add_f32
    }
}

// ---------------------------------------------------------------------------
// Finalize BN stats -> per-column affine (a, c):  bn(h) = h*a + c
// ---------------------------------------------------------------------------
__global__ void bnstats_kernel(const float* __restrict__ colsum,
                               const float* __restrict__ colsumsq,
                               const float* __restrict__ g,
                               const float* __restrict__ beta,
                               float* __restrict__ a, float* __restrict__ c,
                               float invN) {
    int n = threadIdx.x;
    float mu = colsum[n] * invN;
    float var = colsumsq[n] * invN - mu * mu;  // biased variance (training mode)
    float av = g[n] * rsqrtf(var + BN_EPS);
    a[n] = av;
    c[n] = beta[n] - mu * av;
}

// ---------------------------------------------------------------------------
// Final elementwise: out = relu(h2*a2[col] + c2[col])
// ---------------------------------------------------------------------------
__global__ void bnrelu_kernel(const float* __restrict__ h,
                              const float* __restrict__ a,
                              const float* __restrict__ c,
                              float* __restrict__ out, int n) {
    int i = blockIdx.x * blockDim.x + threadIdx.x;
    if (i < n) {
        int col = i & (D - 1);
        out[i] = fmaxf(0.0f, h[i] * a[col] + c[col]);
    }
}

// ---------------------------------------------------------------------------
extern "C" void kernel_launch(void* const* d_in, const int* in_sizes, int n_in,
                              void* d_out, int out_size, void* d_ws, size_t ws_size,
                              hipStream_t stream) {
    const float* x            = (const float*)d_in[0];
    const float* x_bridge     = (const float*)d_in[1];
    const int*   edge_index   = (const int*)d_in[2];   // [2, E]
    const int*   bridge_index = (const int*)d_in[3];   // [E]
    const float* lin_w        = (const float*)d_in[4]; // [128, 256]
    const float* lin_b        = (const float*)d_in[5];
    const float* eps          = (const float*)d_in[6];
    const float* w1           = (const float*)d_in[7];
    const float* b1           = (const float*)d_in[8];
    const float* g1           = (const float*)d_in[9];
    const float* beta1        = (const float*)d_in[10];
    const float* w2           = (const float*)d_in[11];
    const float* b2           = (const float*)d_in[12];
    const float* g2           = (const float*)d_in[13];
    const float* beta2        = (const float*)d_in[14];

    const int N  = in_sizes[0] / D;   // 10000
    const int Nb = in_sizes[1] / D;   // 20000
    const int E  = in_sizes[3];       // 640000

    // workspace layout (floats)
    float* ws  = (float*)d_ws;
    float* xW  = ws;                        // N*D
    float* eW  = xW + (size_t)N * D;        // Nb*D
    float* agg = eW + (size_t)Nb * D;       // N*D
    float* h1  = agg + (size_t)N * D;       // N*D
    float* h2  = h1 + (size_t)N * D;        // N*D
    float* st  = h2 + (size_t)N * D;        // 1024 stats floats
    float *sum1 = st, *sq1 = st + 128, *a1 = st + 256, *c1 = st + 384;
    float *sum2 = st + 512, *sq2 = st + 640, *a2 = st + 768, *c2 = st + 896;

    const int nAgg = N * D;

    // agg = (1+eps)*x ; zero stats accumulators
    init_kernel<<<(nAgg + 255) / 256, 256, 0, stream>>>(x, eps, agg, st, nAgg);

    // per-node transforms: xW = x @ Wa.T ; eW = x_bridge @ Wb.T + lin_b
    gemm_kernel<false, false><<<N / 80, 256, 0, stream>>>(
        x, lin_w, 2 * D, 0, nullptr, nullptr, nullptr, xW, nullptr, nullptr);
    gemm_kernel<false, false><<<Nb / 80, 256, 0, stream>>>(
        x_bridge, lin_w, 2 * D, D, lin_b, nullptr, nullptr, eW, nullptr, nullptr);

    // edge scatter: agg += relu(xW[src] + eW[bridge]) by dst
    long ethreads = (long)E * 32;
    edge_kernel<<<(unsigned)((ethreads + 255) / 256), 256, 0, stream>>>(
        xW, eW, edge_index, edge_index + E, bridge_index, agg, E);

    // layer 1: h1 = agg @ w1.T + b1 (+ column stats)
    gemm_kernel<false, true><<<N / 80, 256, 0, stream>>>(
        agg, w1, D, 0, b1, nullptr, nullptr, h1, sum1, sq1);
    bnstats_kernel<<<1, 128, 0, stream>>>(sum1, sq1, g1, beta1, a1, c1, 1.0f / N);

    // layer 2: h2 = relu(bn1(h1)) @ w2.T + b2 (BN+ReLU fused into A-load)
    gemm_kernel<true, true><<<N / 80, 256, 0, stream>>>(
        h1, w2, D, 0, b2, a1, c1, h2, sum2, sq2);
    bnstats_kernel<<<1, 128, 0, stream>>>(sum2, sq2, g2, beta2, a2, c2, 1.0f / N);

    // final: out = relu(bn2(h2))
    bnrelu_kernel<<<(nAgg + 255) / 256, 256, 0, stream>>>(h2, a2, c2,
                                                          (float*)d_out, nAgg);
}